// GroupEnhance_76690936037555
// MI455X (gfx1250) — compile-verified
//
#include <hip/hip_runtime.h>
#include <hip/hip_bf16.h>

typedef float v2f __attribute__((ext_vector_type(2)));
typedef float v8f __attribute__((ext_vector_type(8)));

#define DIM 128

// ---------------------------------------------------------------------------
// Kernel 0: zero the aggregation buffer (d_out, reused as agg) and deg.
// ---------------------------------------------------------------------------
__global__ __launch_bounds__(256) void zero_kernel(float* __restrict__ agg,
                                                   float* __restrict__ deg,
                                                   int n_agg, int n_deg) {
    int i = blockIdx.x * blockDim.x + threadIdx.x;
    int total = n_agg + n_deg;
    for (; i < total; i += gridDim.x * blockDim.x) {
        if (i < n_agg) agg[i] = 0.0f;
        else           deg[i - n_agg] = 0.0f;
    }
}

// ---------------------------------------------------------------------------
// Kernel 1: wave-per-edge scatter.  msg = x[src] * w  ->  atomic add agg[dst]
// Each lane handles 4 consecutive floats (32 lanes * 4 = 128 = DIM).
// unsafeAtomicAdd lowers to global_atomic_add_f32 (L2-resident accumulator).
// ---------------------------------------------------------------------------
__global__ __launch_bounds__(256) void edge_scatter_kernel(
    const float* __restrict__ x,
    const long long* __restrict__ edge_index,   // [2, E] int64
    const float* __restrict__ edge_weight,
    float* __restrict__ agg,
    float* __restrict__ deg,
    int E) {
    int wave = (blockIdx.x * blockDim.x + threadIdx.x) >> 5;
    int lane = threadIdx.x & 31;
    if (wave >= E) return;

    long long src = edge_index[wave];
    long long dst = edge_index[(long long)E + wave];
    float w = edge_weight[wave];

    const float4* xrow = (const float4*)(x + src * (long long)DIM);
    float4 v = xrow[lane];

    float* arow = agg + dst * (long long)DIM + lane * 4;
    unsafeAtomicAdd(arow + 0, v.x * w);
    unsafeAtomicAdd(arow + 1, v.y * w);
    unsafeAtomicAdd(arow + 2, v.z * w);
    unsafeAtomicAdd(arow + 3, v.w * w);

    if (lane == 0) unsafeAtomicAdd(deg + dst, 1.0f);
}

// ---------------------------------------------------------------------------
// Kernel 2: per 16-row node tile:
//   out = x + alpha * ((agg/deg) @ W^T + b)
// 8 waves/block, each wave owns one 16-col output tile; K swept 4 at a time
// with v_wmma_f32_16x16x4_f32 (exact f32 math).
// A (16xK) tile staged in LDS (stride 132 floats -> no bank conflicts).
// B[k][n] = W[n][k]: per-lane float2 load from row n of W (contiguous).
// In-place on d_out: agg tile is consumed into LDS before the final store.
// ---------------------------------------------------------------------------
__global__ __launch_bounds__(256) void node_gemm_kernel(
    const float* __restrict__ x,
    float* __restrict__ agg_out,          // d_out: agg on entry, result on exit
    const float* __restrict__ deg,
    const float* __restrict__ W,          // [DIM, DIM] row-major (out, in)
    const float* __restrict__ bias,       // [DIM]
    const float* __restrict__ alpha_p,    // scalar
    int N) {
    __shared__ float As[16 * 132];        // padded 16 x 128 tile

    const int tile_m = blockIdx.x * 16;

    // Stage deg-normalized agg tile into LDS.
    for (int i = threadIdx.x; i < 16 * DIM; i += 256) {
        int r = i >> 7;          // 0..15
        int c = i & 127;         // 0..127
        int gm = tile_m + r;
        if (gm >= N) gm = N - 1; // clamp (N % 16 == 0 for this problem)
        float dg = deg[gm];
        if (dg < 1.0f) dg = 1.0f;
        As[r * 132 + c] = agg_out[(long long)gm * DIM + c] / dg;
    }
    __syncthreads();

    const int lane = threadIdx.x & 31;
    const int n0   = (threadIdx.x >> 5) * 16;  // wave's output-column base
    const int m    = lane & 15;                // A row / B col / C col
    const int kh   = lane >> 4;                // K-half select (0 or 1)

    v8f acc = {0.f, 0.f, 0.f, 0.f, 0.f, 0.f, 0.f, 0.f};

    const float* wrow = W + (long long)(n0 + m) * DIM;   // row n of W = col n of B

    for (int k0 = 0; k0 < DIM; k0 += 4) {
        int kb = k0 + 2 * kh;
        // A fragment: A[m][kb], A[m][kb+1]   (ds_load_b64, conflict-free)
        v2f a = *(const v2f*)(&As[m * 132 + kb]);
        // B fragment: B[kb][n] = W[n][kb], W[n][kb+1]
        v2f b = *(const v2f*)(&wrow[kb]);
        acc = __builtin_amdgcn_wmma_f32_16x16x4_f32(
            /*neg_a=*/false, a, /*neg_b=*/false, b,
            /*c_mod=*/(short)0, acc, /*reuse_a=*/false, /*reuse_b=*/false);
    }

    // Epilogue: C/D layout -> VGPR r holds (row = r + 8*kh, col = lane&15).
    const float alpha = *alpha_p;
    const int gc = n0 + m;
    const float bv = bias[gc];
    #pragma unroll
    for (int r = 0; r < 8; ++r) {
        int gm = tile_m + r + 8 * kh;
        if (gm < N) {
            float val = x[(long long)gm * DIM + gc] + alpha * (acc[r] + bv);
            agg_out[(long long)gm * DIM + gc] = val;
        }
    }
}

// ---------------------------------------------------------------------------
extern "C" void kernel_launch(void* const* d_in, const int* in_sizes, int n_in,
                              void* d_out, int out_size, void* d_ws, size_t ws_size,
                              hipStream_t stream) {
    const float*     x   = (const float*)d_in[0];
    const long long* ei  = (const long long*)d_in[1];   // int64 [2, E]
    const float*     ew  = (const float*)d_in[3];
    const float*     W   = (const float*)d_in[4];
    const float*     b   = (const float*)d_in[5];
    const float*     alp = (const float*)d_in[6];

    const int N = in_sizes[0] / DIM;   // 100000
    const int E = in_sizes[3];         // 640000

    float* agg = (float*)d_out;        // reuse output buffer as accumulator
    float* deg = (float*)d_ws;         // N floats of workspace

    // 1) zero agg + deg
    {
        int total = N * DIM + N;
        int blocks = (total + 255) / 256;
        if (blocks > 16384) blocks = 16384;
        zero_kernel<<<blocks, 256, 0, stream>>>(agg, deg, N * DIM, N);
    }

    // 2) edge scatter (one wave32 per edge, 8 waves per block)
    {
        int blocks = (E + 7) / 8;
        edge_scatter_kernel<<<blocks, 256, 0, stream>>>(x, ei, ew, agg, deg, E);
    }

    // 3) normalize + GEMM (WMMA f32) + bias + residual, in place on d_out
    {
        int blocks = (N + 15) / 16;
        node_gemm_kernel<<<blocks, 256, 0, stream>>>(x, agg, deg, W, b, alp, N);
    }
}